// Cell_First_6631429505478
// MI455X (gfx1250) — compile-verified
//
#include <hip/hip_runtime.h>

typedef __attribute__((ext_vector_type(2))) float v2f;
typedef __attribute__((ext_vector_type(8))) float v8f;

#define D 128
#define LDS_STRIDE 132  // pad 16x128 tile rows -> conflict-free column access (528B = 16B aligned)

// ---------------------------------------------------------------- utilities
__global__ void k_fill_zero(float* __restrict__ p, long long n) {
    long long i = (long long)blockIdx.x * blockDim.x + threadIdx.x;
    if (i < n) p[i] = 0.0f;
}

__global__ void k_deg(const int* __restrict__ dst, float* __restrict__ deg, int E) {
    int e = blockIdx.x * blockDim.x + threadIdx.x;
    if (e < E) atomicAdd(&deg[dst[e]], 1.0f);
}

__global__ void k_invdeg(float* __restrict__ deg, int N) {
    int i = blockIdx.x * blockDim.x + threadIdx.x;
    if (i < N) { float d = deg[i]; deg[i] = 1.0f / fmaxf(d, 1.0f); }
}

// one wave32 per edge: 32 lanes x float4 = 128 floats
__global__ void k_scatter(const float* __restrict__ x, const int* __restrict__ src,
                          const int* __restrict__ dst, float* __restrict__ agg, int E) {
    int gid = blockIdx.x * blockDim.x + threadIdx.x;
    int e = gid >> 5;
    int lane = gid & 31;
    if (e >= E) return;
    int s = src[e], d = dst[e];
    const float4 v = *(const float4*)(x + (long long)s * D + lane * 4);
    float* base = agg + (long long)d * D + lane * 4;
    atomicAdd(base + 0, v.x);
    atomicAdd(base + 1, v.y);
    atomicAdd(base + 2, v.z);
    atomicAdd(base + 3, v.w);
}

__global__ void k_scale_rows(float* __restrict__ agg, const float* __restrict__ inv_deg, int N) {
    int gid = blockIdx.x * blockDim.x + threadIdx.x;  // N*32 threads, float4 each
    if (gid >= N * 32) return;
    int row = gid >> 5, c4 = (gid & 31) * 4;
    float s = inv_deg[row];
    float4* p = (float4*)(agg + (long long)row * D + c4);
    float4 v = *p;
    v.x *= s; v.y *= s; v.z *= s; v.w *= s;
    *p = v;
}

// ------------------------------------------------- fused GEMM (z = X*W^T + b) + BN stats
// block = 256 threads = 8 waves; block covers rows [m0, m0+16), each wave one 16-col tile.
__global__ void __launch_bounds__(256)
k_gemm_stats(const float* __restrict__ X, const float* __restrict__ Wm,
             const float* __restrict__ bias, float* __restrict__ Z,
             float* __restrict__ colsum, float* __restrict__ colsumsq, int Nrows) {
    __shared__ float sA[16 * LDS_STRIDE];
    const int t = threadIdx.x;
    const int m0 = blockIdx.x * 16;
    const int wave = t >> 5;
    const int lane = t & 31;
    const int o0 = wave * 16;

    // ---- stage X[m0:m0+16, 0:128] into LDS via CDNA5 async direct-to-LDS copy
    // (VDST VGPR = wave-relative LDS byte offset; VADDR = 64-bit global address)
#pragma unroll
    for (int it = 0; it < 2; ++it) {
        int chunk = t + it * 256;           // 0..511 chunks of 16B
        int r = chunk >> 5;                 // 0..15
        int kb = (chunk & 31) * 4;          // 0..124
        int row = m0 + r;
        if (row >= Nrows) row = Nrows - 1;  // clamp; garbage rows masked at store
        unsigned lds_addr = (unsigned)(size_t)(sA + r * LDS_STRIDE + kb);
        unsigned long long gaddr =
            (unsigned long long)(size_t)(X + (size_t)row * D + kb);
        asm volatile("global_load_async_to_lds_b128 %0, %1, off"
                     :: "v"(lds_addr), "v"(gaddr)
                     : "memory");
    }
    asm volatile("s_wait_asynccnt 0x0" ::: "memory");
    __syncthreads();

    const int halfs = lane >> 4;            // 0: lanes 0-15, 1: lanes 16-31
    const int l15 = lane & 15;
    const int koff = halfs * 2;             // A/B lane halves carry K pairs {0,1} / {2,3}
    const float* wrow = Wm + (long long)(o0 + l15) * D;   // B[k][n] = W[n][k]
    const float* arow = sA + l15 * LDS_STRIDE;

    v8f acc = {};
#pragma unroll
    for (int k0 = 0; k0 < D; k0 += 4) {
        v2f a, b;
        a.x = arow[k0 + koff];
        a.y = arow[k0 + koff + 1];
        b.x = wrow[k0 + koff];
        b.y = wrow[k0 + koff + 1];
        acc = __builtin_amdgcn_wmma_f32_16x16x4_f32(
            /*neg_a=*/false, a, /*neg_b=*/false, b,
            /*c_mod=*/(short)0, acc, /*reuse_a=*/false, /*reuse_b=*/false);
    }

    // C/D layout: VGPR g -> (M = g + 8*halfs, N = lane&15)
    const int col = o0 + l15;
    const float bv = bias[col];
    const int rowbase = m0 + halfs * 8;
    float s1 = 0.f, s2 = 0.f;
    if (m0 + 16 <= Nrows) {
        // fast path: full tile, no exec-mask diamonds
#pragma unroll
        for (int g = 0; g < 8; ++g) {
            float v = acc[g] + bv;
            Z[(long long)(rowbase + g) * D + col] = v;
            s1 += v;
            s2 += v * v;
        }
    } else {
#pragma unroll
        for (int g = 0; g < 8; ++g) {
            int row = rowbase + g;
            if (row < Nrows) {
                float v = acc[g] + bv;
                Z[(long long)row * D + col] = v;
                s1 += v;
                s2 += v * v;
            }
        }
    }
    // fold the two M-halves of the same column, one atomic per column per wave
    s1 += __shfl_xor(s1, 16);
    s2 += __shfl_xor(s2, 16);
    if (halfs == 0) {
        atomicAdd(&colsum[col], s1);
        atomicAdd(&colsumsq[col], s2);
    }
}

// ------------------------------------------------- BN + ReLU + arch-weight mix, accumulate
__device__ __forceinline__ float bn_relu(float z, float su, float sq, float invN,
                                         float g, float bt) {
    float mu = su * invN;
    float var = sq * invN - mu * mu;
    float rs = rsqrtf(var + 1e-5f);
    float v = (z - mu) * rs * g + bt;
    return v > 0.f ? v : 0.f;
}

__global__ void k_bn_apply(const float* __restrict__ Z,      // [3][N][128]
                           const float* __restrict__ stats,  // [3]{sum[128], sumsq[128]}
                           const float* __restrict__ wts,    // [3]
                           const float* __restrict__ gamma,  // [3][128]
                           const float* __restrict__ beta,   // [3][128]
                           float* __restrict__ outp, int Nrows) {
    int gid = blockIdx.x * blockDim.x + threadIdx.x;
    if (gid >= Nrows * 32) return;
    int row = gid >> 5, c4 = (gid & 31) * 4;
    const float invN = 1.0f / (float)Nrows;
    float ax = 0.f, ay = 0.f, az = 0.f, aw = 0.f;
#pragma unroll
    for (int c = 0; c < 3; ++c) {
        const float4 z  = *(const float4*)(Z + ((long long)c * Nrows + row) * D + c4);
        const float4 su = *(const float4*)(stats + c * 256 + c4);
        const float4 sq = *(const float4*)(stats + c * 256 + 128 + c4);
        const float4 g  = *(const float4*)(gamma + c * D + c4);
        const float4 bt = *(const float4*)(beta + c * D + c4);
        const float w = wts[c];
        ax += w * bn_relu(z.x, su.x, sq.x, invN, g.x, bt.x);
        ay += w * bn_relu(z.y, su.y, sq.y, invN, g.y, bt.y);
        az += w * bn_relu(z.z, su.z, sq.z, invN, g.z, bt.z);
        aw += w * bn_relu(z.w, su.w, sq.w, invN, g.w, bt.w);
    }
    float4* po = (float4*)(outp + (long long)row * D + c4);
    float4 o = *po;
    o.x += ax; o.y += ay; o.z += az; o.w += aw;
    *po = o;
}

// ---------------------------------------------------------------- host driver
extern "C" void kernel_launch(void* const* d_in, const int* in_sizes, int n_in,
                              void* d_out, int out_size, void* d_ws, size_t ws_size,
                              hipStream_t stream) {
    const int*   edge_index = (const int*)d_in[0];
    const float* h       = (const float*)d_in[1];
    const float* h_in    = (const float*)d_in[2];
    const float* weights = (const float*)d_in[3];   // [6][3]
    const float* Wall    = (const float*)d_in[4];   // [6][3][128][128]
    const float* ball    = (const float*)d_in[5];   // [6][3][128]
    const float* gall    = (const float*)d_in[6];   // [6][3][128]
    const float* btall   = (const float*)d_in[7];   // [6][3][128]
    float* out = (float*)d_out;                     // [3][N][128]

    const int N = in_sizes[1] / D;
    const int E = in_sizes[0] / 2;
    const int* src = edge_index;
    const int* dst = edge_index + E;

    float* ws    = (float*)d_ws;
    float* deg   = ws;                                   // N
    float* aggH  = deg   + (size_t)N;                    // N*128
    float* aggS1 = aggH  + (size_t)N * D;                // N*128
    float* aggS2 = aggS1 + (size_t)N * D;                // N*128
    float* zbuf  = aggS2 + (size_t)N * D;                // 3*N*128
    float* stats = zbuf  + (size_t)3 * N * D;            // 3*256

    float* s1 = out;
    float* s2 = out + (size_t)N * D;
    float* s3 = out + (size_t)2 * N * D;

    const int TB = 256;
    auto blocks = [](long long n) { return (unsigned)((n + 255) / 256); };
    const unsigned gElem  = blocks((long long)N * 32);   // float4-per-thread grids
    const unsigned gGemm  = (unsigned)((N + 15) / 16);

    // degree -> inverse degree
    k_fill_zero<<<blocks(N), TB, 0, stream>>>(deg, N);
    k_fill_zero<<<blocks((long long)3 * N * D), TB, 0, stream>>>(out, (long long)3 * N * D);
    k_deg<<<blocks(E), TB, 0, stream>>>(dst, deg, E);
    k_invdeg<<<blocks(N), TB, 0, stream>>>(deg, N);

    auto run_agg = [&](const float* x, float* agg) {
        k_fill_zero<<<blocks((long long)N * D), TB, 0, stream>>>(agg, (long long)N * D);
        k_scatter<<<blocks((long long)E * 32), TB, 0, stream>>>(x, src, dst, agg, E);
        k_scale_rows<<<gElem, TB, 0, stream>>>(agg, deg, N);
    };

    auto run_mixed = [&](int m, const float* x, const float* aggx, float* outstate) {
        k_fill_zero<<<3, TB, 0, stream>>>(stats, 768);
        const float* branches[3] = { aggx, x, h_in };
        for (int c = 0; c < 3; ++c) {
            const size_t mc = (size_t)m * 3 + c;
            k_gemm_stats<<<gGemm, TB, 0, stream>>>(
                branches[c], Wall + mc * D * D, ball + mc * D,
                zbuf + (size_t)c * N * D, stats + c * 256, stats + c * 256 + 128, N);
        }
        k_bn_apply<<<gElem, TB, 0, stream>>>(
            zbuf, stats, weights + (size_t)m * 3,
            gall + (size_t)m * 3 * D, btall + (size_t)m * 3 * D, outstate, N);
    };

    // step 1: s1 = mixed(0, h)
    run_agg(h, aggH);
    run_mixed(0, h, aggH, s1);
    // step 2: s2 = mixed(1, h) + mixed(2, s1)
    run_agg(s1, aggS1);
    run_mixed(1, h, aggH, s2);
    run_mixed(2, s1, aggS1, s2);
    // step 3: s3 = mixed(3, h) + mixed(4, s1) + mixed(5, s2)
    run_agg(s2, aggS2);
    run_mixed(3, h, aggH, s3);
    run_mixed(4, s1, aggS1, s3);
    run_mixed(5, s2, aggS2, s3);
}